// SlotAttention_11252814316138
// MI455X (gfx1250) — compile-verified
//
#include <hip/hip_runtime.h>
#include <hip/hip_bf16.h>

// ---------------------------------------------------------------------------
// Slot Attention fused pipeline for gfx1250 (CDNA5, wave32, WMMA f32 16x16x4)
// ---------------------------------------------------------------------------
#define Bn   32
#define Nn   4096
#define Cn   128
#define Dn   64
#define Kn   8
#define Hn   128
#define SCALE 0.125f      // 64^-0.5
#define EPS   1e-8f

typedef __attribute__((ext_vector_type(2))) float v2f;
typedef __attribute__((ext_vector_type(8))) float v8f;

static __device__ __forceinline__ v8f wmma4(v2f a, v2f b, v8f c) {
  // D = A(16x4) * B(4x16) + C(16x16), all f32
  return __builtin_amdgcn_wmma_f32_16x16x4_f32(false, a, false, b,
                                               (short)0, c, false, false);
}

// ---------------------------------------------------------------------------
// Kernel 1: x = LN(feat); k = x@Wk; v = x@Wv     (streams feat exactly once)
// 64 rows per block, 8 waves, WMMA over K=128 in steps of 4.
// ---------------------------------------------------------------------------
#define ROWS1 64
#define XSTR  132   // 128 + 4 pad; 132*4B row stride is a 16B multiple

__global__ __launch_bounds__(256) void ln_kv_kernel(
    const float* __restrict__ feat, const float* __restrict__ gamma,
    const float* __restrict__ beta, const float* __restrict__ Wk,
    const float* __restrict__ Wv, float* __restrict__ kout,
    float* __restrict__ vout)
{
  __shared__ __align__(16) float xs[ROWS1][XSTR];
  __shared__ float red_s[256], red_q[256];
  __shared__ float mrow[ROWS1], irow[ROWS1];
  const int t = threadIdx.x;
  const long base = (long)blockIdx.x * ROWS1;

  // float4 staged load: 2048 vec4 / 256 threads = 8 per thread (global_load_b128)
  #pragma unroll
  for (int i = 0; i < 8; ++i) {
    int idx = t + i * 256;                 // vec4 index
    int r = idx >> 5, c4 = (idx & 31) << 2;
    *(float4*)&xs[r][c4] = *(const float4*)&feat[(base + r) * Cn + c4];
  }
  __syncthreads();

  { // LayerNorm stats: 4 threads per row
    int r = t >> 2, q = t & 3;
    float s = 0.f, ss = 0.f;
    for (int c = q * 32; c < q * 32 + 32; ++c) { float x = xs[r][c]; s += x; ss += x * x; }
    red_s[t] = s; red_q[t] = ss;
  }
  __syncthreads();
  if ((t & 3) == 0) {
    int r = t >> 2;
    float s  = red_s[t] + red_s[t + 1] + red_s[t + 2] + red_s[t + 3];
    float ss = red_q[t] + red_q[t + 1] + red_q[t + 2] + red_q[t + 3];
    float m = s * (1.f / 128.f);
    float var = ss * (1.f / 128.f) - m * m;
    mrow[r] = m; irow[r] = rsqrtf(var + 1e-5f);
  }
  __syncthreads();
  #pragma unroll
  for (int i = 0; i < (ROWS1 * Cn) / 256; ++i) {
    int idx = t + i * 256;
    int r = idx >> 7, c = idx & 127;
    xs[r][c] = (xs[r][c] - mrow[r]) * irow[r] * gamma[c] + beta[c];
  }
  __syncthreads();

  // WMMA phase: jobs = mat(2) x row_tile(4) x col_tile(4) = 32, 4 per wave
  const int lane  = t & 31;
  const int wv    = t >> 5;
  const int lmod  = lane & 15;
  const int khalf = (lane >> 4) << 1;          // K offset: 0 (lanes<16), 2 (>=16)
  const int rowhi = (lane >> 4) << 3;          // C row offset: +8 for upper lanes
  for (int j = 0; j < 4; ++j) {
    int job = wv + (j << 3);
    int mat = job >> 4;
    int rt  = (job >> 2) & 3;
    int ct  = job & 3;
    const float* W = mat ? Wv : Wk;
    // hoist lane-dependent address part: all loop loads become base+immediate
    const float* Wbase = W + (size_t)(khalf * Dn + ct * 16 + lmod);
    v8f acc = {};
    #pragma unroll
    for (int kk = 0; kk < Cn; kk += 4) {
      v2f a, bb;
      a.x  = xs[rt * 16 + lmod][kk + khalf];
      a.y  = xs[rt * 16 + lmod][kk + khalf + 1];
      bb.x = Wbase[kk * Dn];                 // K = kk + khalf
      bb.y = Wbase[(kk + 1) * Dn];           // K = kk + khalf + 1
      acc = wmma4(a, bb, acc);
    }
    float* dst = mat ? vout : kout;
    #pragma unroll
    for (int r = 0; r < 8; ++r) {
      long row = base + rt * 16 + r + rowhi;
      dst[row * Dn + ct * 16 + lmod] = acc[r];
    }
  }
}

// ---------------------------------------------------------------------------
// Kernel: copy initial slots into the iteration buffer
// ---------------------------------------------------------------------------
__global__ __launch_bounds__(256) void copy_slots_kernel(
    const float* __restrict__ src, float* __restrict__ dst)
{
  int i = blockIdx.x * 256 + threadIdx.x;
  if (i < Bn * Kn * Dn) dst[i] = src[i];
}

// ---------------------------------------------------------------------------
// Kernel 2 (per iter): q = LN(slot) @ Wq (bg row 0 uses its own LN/W);
// pads q rows 8..15 with zeros; zeroes the accumulators for this iteration.
// ---------------------------------------------------------------------------
__global__ __launch_bounds__(256) void qprep_kernel(
    const float* __restrict__ slotbuf,
    const float* __restrict__ q_g,   const float* __restrict__ q_b,
    const float* __restrict__ qbg_g, const float* __restrict__ qbg_b,
    const float* __restrict__ Wq,    const float* __restrict__ Wq_bg,
    float* __restrict__ qout, float* __restrict__ gacc, float* __restrict__ gsum)
{
  __shared__ float sl[8][64];
  __shared__ float lnv[8][64];
  __shared__ float mS[8], iS[8];
  const int t = threadIdx.x;
  const int b = blockIdx.x;

  for (int i = t; i < 512; i += 256) sl[i >> 6][i & 63] = slotbuf[b * 512 + i];
  __syncthreads();
  if (t < 8) {
    float s = 0.f, ss = 0.f;
    for (int d = 0; d < 64; ++d) { float x = sl[t][d]; s += x; ss += x * x; }
    float m = s * (1.f / 64.f);
    float var = ss * (1.f / 64.f) - m * m;
    mS[t] = m; iS[t] = rsqrtf(var + 1e-5f);
  }
  __syncthreads();
  for (int i = t; i < 512; i += 256) {
    int r = i >> 6, d = i & 63;
    const float* gg = (r == 0) ? qbg_g : q_g;
    const float* bb = (r == 0) ? qbg_b : q_b;
    lnv[r][d] = (sl[r][d] - mS[r]) * iS[r] * gg[d] + bb[d];
  }
  __syncthreads();
  for (int i = t; i < 512; i += 256) {
    int r = i >> 6, j = i & 63;
    const float* W = (r == 0) ? Wq_bg : Wq;
    float acc = 0.f;
    for (int d = 0; d < 64; ++d) acc += lnv[r][d] * W[d * 64 + j];
    qout[b * 1024 + r * 64 + j] = acc;
  }
  for (int i = t; i < 512;  i += 256) qout[b * 1024 + 512 + i] = 0.f; // pad rows
  for (int i = t; i < 1024; i += 256) gacc[b * 1024 + i] = 0.f;
  if (t < 16) gsum[b * 16 + t] = 0.f;
}

// ---------------------------------------------------------------------------
// Kernel 3 (per iter): fused dots -> column softmax (slot axis) -> attn@v
// accumulation and attn row-sums. One pass over k and v.
// grid (B, N/256); each wave owns 32 columns (two 16-wide WMMA tiles).
// ---------------------------------------------------------------------------
__global__ __launch_bounds__(256) void attn_kernel(
    const float* __restrict__ q, const float* __restrict__ k,
    const float* __restrict__ v, float* __restrict__ gacc,
    float* __restrict__ gsum, float* __restrict__ attn_out, int write_attn)
{
  __shared__ __align__(16) float qs[16][68];   // 68*4B stride = 16B multiple
  __shared__ float att[8][16][36];             // per-wave 16x32 attn tile (+pad)
  __shared__ float accred[16][64];
  __shared__ float sumred[16];
  const int t = threadIdx.x;
  const int b = blockIdx.x;
  const int lane  = t & 31;
  const int wv    = t >> 5;
  const int lmod  = lane & 15;
  const int khalf = (lane >> 4) << 1;
  const int rowbase = (lane >> 4) << 3;

  { // q stage: 256 float4 = one b128 load per thread
    int r = t >> 4, c4 = (t & 15) << 2;
    *(float4*)&qs[r][c4] = *(const float4*)&q[b * 1024 + t * 4];
  }
  for (int i = t; i < 1024; i += 256) accred[i >> 6][i & 63] = 0.f;
  if (t < 16) sumred[t] = 0.f;
  __syncthreads();

  const int colbase = blockIdx.y * 256 + wv * 32;
  const float* kb = k + ((long)b * Nn + colbase) * Dn;
  const float* vb = v + ((long)b * Nn + colbase) * Dn;
  __builtin_prefetch(vb, 0, 1);         // warm v stream (global_prefetch_b8)

  // per-lane base pointers (contiguous K pairs -> single b64 loads)
  const float* k0 = kb + lmod * Dn + khalf;
  const float* k1 = kb + (16 + lmod) * Dn + khalf;

  // dots[16 slots x 32 cols] = q @ k^T
  v8f cd[2] = {};
  #pragma unroll
  for (int kk = 0; kk < Dn; kk += 4) {
    v2f a, b0, b1;
    a.x = qs[lmod][kk + khalf];
    a.y = qs[lmod][kk + khalf + 1];
    b0  = *(const v2f*)(k0 + kk);       // global_load_b64, base+imm
    b1  = *(const v2f*)(k1 + kk);
    cd[0] = wmma4(a, b0, cd[0]);
    cd[1] = wmma4(a, b1, cd[1]);
  }

  // softmax over the 8 slots: all 8 values of a column live in one lane
  float srow[8];
  #pragma unroll
  for (int r = 0; r < 8; ++r) srow[r] = 0.f;
  #pragma unroll
  for (int tile = 0; tile < 2; ++tile) {
    float av[8];
    if (lane < 16) {
      float m = -1e30f;
      #pragma unroll
      for (int r = 0; r < 8; ++r) m = fmaxf(m, cd[tile][r] * SCALE);
      float s = 0.f;
      #pragma unroll
      for (int r = 0; r < 8; ++r) { float e = __expf(cd[tile][r] * SCALE - m); av[r] = e; s += e; }
      float inv = 1.f / s;
      #pragma unroll
      for (int r = 0; r < 8; ++r) av[r] = av[r] * inv + EPS;
    } else {
      #pragma unroll
      for (int r = 0; r < 8; ++r) av[r] = 0.f;   // zero pad rows 8..15
    }
    int cc = tile * 16 + lmod;
    #pragma unroll
    for (int r = 0; r < 8; ++r) att[wv][rowbase + r][cc] = av[r];
    if (lane < 16) {
      #pragma unroll
      for (int r = 0; r < 8; ++r) srow[r] += av[r];
      if (write_attn) {
        #pragma unroll
        for (int r = 0; r < 8; ++r)
          attn_out[((long)b * Kn + r) * Nn + colbase + cc] = av[r];
      }
    }
  }
  // per-slot attn sums over this wave's 32 columns -> block accumulator
  #pragma unroll
  for (int r = 0; r < 8; ++r) {
    float s = srow[r];
    for (int off = 16; off > 0; off >>= 1) s += __shfl_xor(s, off, 32);
    if (lane == 0) atomicAdd(&sumred[r], s);
  }
  __syncthreads();

  // acc[16 slots x 64 d] += attn(16x32) @ v(32x64)
  const float* v0 = vb + khalf * Dn + lmod;     // lane-hoisted base
  v8f acc[4] = {};
  #pragma unroll
  for (int kk = 0; kk < 32; kk += 4) {
    v2f a;
    a.x = att[wv][lmod][kk + khalf];
    a.y = att[wv][lmod][kk + khalf + 1];
    #pragma unroll
    for (int dt = 0; dt < 4; ++dt) {
      v2f bb;
      bb.x = v0[kk * Dn + dt * 16];             // K = kk + khalf
      bb.y = v0[(kk + 1) * Dn + dt * 16];       // K = kk + khalf + 1
      acc[dt] = wmma4(a, bb, acc[dt]);
    }
  }
  #pragma unroll
  for (int dt = 0; dt < 4; ++dt)
    #pragma unroll
    for (int r = 0; r < 8; ++r)
      atomicAdd(&accred[rowbase + r][dt * 16 + lmod], acc[dt][r]);
  __syncthreads();

  for (int i = t; i < 1024; i += 256)
    atomicAdd(&gacc[b * 1024 + i], accred[i >> 6][i & 63]);
  if (t < 16) atomicAdd(&gsum[b * 16 + t], sumred[t]);
}

// ---------------------------------------------------------------------------
// Kernel 4 (per iter): upd = acc/sum; GRU; residual MLP. One wave per slot.
// ---------------------------------------------------------------------------
__global__ __launch_bounds__(256) void slot_update_kernel(
    const float* __restrict__ gacc, const float* __restrict__ gsum,
    float* __restrict__ slotbuf,
    const float* __restrict__ gru_wih, const float* __restrict__ gru_whh,
    const float* __restrict__ gru_bih, const float* __restrict__ gru_bhh,
    const float* __restrict__ gbg_wih, const float* __restrict__ gbg_whh,
    const float* __restrict__ gbg_bih, const float* __restrict__ gbg_bhh,
    const float* __restrict__ rln_g, const float* __restrict__ rln_b,
    const float* __restrict__ rw1, const float* __restrict__ rb1,
    const float* __restrict__ rw2, const float* __restrict__ rb2,
    const float* __restrict__ rbg_ln_g, const float* __restrict__ rbg_ln_b,
    const float* __restrict__ rbg_w1, const float* __restrict__ rbg_b1,
    const float* __restrict__ rbg_w2, const float* __restrict__ rbg_b2,
    float* __restrict__ out_slots, int last)
{
  __shared__ float upd_s[8][64], h_s[8][64];
  __shared__ float gi_s[8][192], gh_s[8][192];
  __shared__ float ns_s[8][64], ln_s[8][64], h1_s[8][128];
  const int b = blockIdx.x, t = threadIdx.x;
  const int lane = t & 31, w = t >> 5;
  const int bg = (w == 0);
  const float* wih = bg ? gbg_wih : gru_wih;
  const float* whh = bg ? gbg_whh : gru_whh;
  const float* bih = bg ? gbg_bih : gru_bih;
  const float* bhh = bg ? gbg_bhh : gru_bhh;

  float invsum = 1.f / gsum[b * 16 + w];
  for (int e = lane; e < 64; e += 32) {
    upd_s[w][e] = gacc[b * 1024 + w * 64 + e] * invsum;
    h_s[w][e]   = slotbuf[b * 512 + w * 64 + e];
  }
  __syncthreads();
  for (int g = lane; g < 192; g += 32) {
    float a0 = bih[g], a1 = bhh[g];
    for (int d = 0; d < 64; ++d) {
      a0 += upd_s[w][d] * wih[g * 64 + d];
      a1 += h_s[w][d]   * whh[g * 64 + d];
    }
    gi_s[w][g] = a0; gh_s[w][g] = a1;
  }
  __syncthreads();
  for (int e = lane; e < 64; e += 32) {
    float r = 1.f / (1.f + __expf(-(gi_s[w][e]       + gh_s[w][e])));
    float z = 1.f / (1.f + __expf(-(gi_s[w][64 + e]  + gh_s[w][64 + e])));
    float n = tanhf(gi_s[w][128 + e] + r * gh_s[w][128 + e]);
    ns_s[w][e] = (1.f - z) * n + z * h_s[w][e];
  }
  __syncthreads();
  // LayerNorm over the 64-dim slot (wave reduction)
  float x0 = ns_s[w][lane], x1 = ns_s[w][lane + 32];
  float s = x0 + x1, ss = x0 * x0 + x1 * x1;
  for (int off = 16; off > 0; off >>= 1) {
    s  += __shfl_xor(s,  off, 32);
    ss += __shfl_xor(ss, off, 32);
  }
  float m = s * (1.f / 64.f);
  float var = ss * (1.f / 64.f) - m * m;
  float inv = rsqrtf(var + 1e-5f);
  const float* lg = bg ? rbg_ln_g : rln_g;
  const float* lb = bg ? rbg_ln_b : rln_b;
  for (int e = lane; e < 64; e += 32) ln_s[w][e] = (ns_s[w][e] - m) * inv * lg[e] + lb[e];
  __syncthreads();
  const float* w1 = bg ? rbg_w1 : rw1;
  const float* b1 = bg ? rbg_b1 : rb1;
  for (int j = lane; j < 128; j += 32) {
    float a = b1[j];
    for (int d = 0; d < 64; ++d) a += ln_s[w][d] * w1[d * 128 + j];
    h1_s[w][j] = fmaxf(a, 0.f);
  }
  __syncthreads();
  const float* w2 = bg ? rbg_w2 : rw2;
  const float* b2 = bg ? rbg_b2 : rb2;
  for (int e = lane; e < 64; e += 32) {
    float o = b2[e];
    for (int j = 0; j < 128; ++j) o += h1_s[w][j] * w2[j * 64 + e];
    float val = ns_s[w][e] + o;
    slotbuf[b * 512 + w * 64 + e] = val;
    if (last) out_slots[b * 512 + w * 64 + e] = val;
  }
}

// ---------------------------------------------------------------------------
extern "C" void kernel_launch(void* const* d_in, const int* in_sizes, int n_in,
                              void* d_out, int out_size, void* d_ws, size_t ws_size,
                              hipStream_t stream) {
  const float* feat     = (const float*)d_in[0];
  const float* slot_in  = (const float*)d_in[1];
  const float* nfg      = (const float*)d_in[2];
  const float* nfb      = (const float*)d_in[3];
  const float* Wk       = (const float*)d_in[4];
  const float* Wv       = (const float*)d_in[5];
  const float* q_ln_g   = (const float*)d_in[6];
  const float* q_ln_b   = (const float*)d_in[7];
  const float* Wq       = (const float*)d_in[8];
  const float* qbg_ln_g = (const float*)d_in[9];
  const float* qbg_ln_b = (const float*)d_in[10];
  const float* Wq_bg    = (const float*)d_in[11];
  const float* gru_wih  = (const float*)d_in[12];
  const float* gru_whh  = (const float*)d_in[13];
  const float* gru_bih  = (const float*)d_in[14];
  const float* gru_bhh  = (const float*)d_in[15];
  const float* gbg_wih  = (const float*)d_in[16];
  const float* gbg_whh  = (const float*)d_in[17];
  const float* gbg_bih  = (const float*)d_in[18];
  const float* gbg_bhh  = (const float*)d_in[19];
  const float* rln_g    = (const float*)d_in[20];
  const float* rln_b    = (const float*)d_in[21];
  const float* rw1      = (const float*)d_in[22];
  const float* rb1      = (const float*)d_in[23];
  const float* rw2      = (const float*)d_in[24];
  const float* rb2      = (const float*)d_in[25];
  const float* rbg_ln_g = (const float*)d_in[26];
  const float* rbg_ln_b = (const float*)d_in[27];
  const float* rbg_w1   = (const float*)d_in[28];
  const float* rbg_b1   = (const float*)d_in[29];
  const float* rbg_w2   = (const float*)d_in[30];
  const float* rbg_b2   = (const float*)d_in[31];

  float* out = (float*)d_out;          // [0,16384): slots ; [16384,...): attn
  float* ws  = (float*)d_ws;
  float* kbuf    = ws;                                    // B*N*D
  float* vbuf    = kbuf + (size_t)Bn * Nn * Dn;           // B*N*D
  float* slotbuf = vbuf + (size_t)Bn * Nn * Dn;           // B*8*64
  float* qbuf    = slotbuf + Bn * Kn * Dn;                // B*16*64
  float* gacc    = qbuf + Bn * 16 * 64;                   // B*16*64
  float* gsum    = gacc + Bn * 16 * 64;                   // B*16

  ln_kv_kernel<<<(Bn * Nn) / ROWS1, 256, 0, stream>>>(feat, nfg, nfb, Wk, Wv,
                                                      kbuf, vbuf);
  copy_slots_kernel<<<(Bn * Kn * Dn + 255) / 256, 256, 0, stream>>>(slot_in, slotbuf);

  for (int it = 0; it < 3; ++it) {
    int lastIt = (it == 2);
    qprep_kernel<<<Bn, 256, 0, stream>>>(slotbuf, q_ln_g, q_ln_b, qbg_ln_g,
                                         qbg_ln_b, Wq, Wq_bg, qbuf, gacc, gsum);
    attn_kernel<<<dim3(Bn, Nn / 256), 256, 0, stream>>>(qbuf, kbuf, vbuf, gacc,
                                                        gsum, out + Bn * Kn * Dn,
                                                        lastIt);
    slot_update_kernel<<<Bn, 256, 0, stream>>>(gacc, gsum, slotbuf,
        gru_wih, gru_whh, gru_bih, gru_bhh,
        gbg_wih, gbg_whh, gbg_bih, gbg_bhh,
        rln_g, rln_b, rw1, rb1, rw2, rb2,
        rbg_ln_g, rbg_ln_b, rbg_w1, rbg_b1, rbg_w2, rbg_b2,
        out, lastIt);
  }
}